// radiusgraphEdge_85787676770519
// MI455X (gfx1250) — compile-verified
//
#include <hip/hip_runtime.h>

// radius-graph edge kernel for MI455X (gfx1250).
// dist[i][j] = || minimum_image(x[j] - x[i]) ||  (L = 1 periodic box)
// mask[i][j] = (dist < 0.1) && (i != j), written as 1.0f / 0.0f
// d_out = [ dist (N*N floats) | mask (N*N floats) ]
//
// Store-bandwidth-bound: 2*N^2*4B = 512 MiB @ 23.3 TB/s ~= 23 us floor.
// CDNA5 paths used: global_load_async_to_lds_b128 (ASYNCcnt) for the column
// position tile, ds_load_b128 readback, nontemporal b128 output stores.

#define THREADS        256
#define TILE_COLS      1024   // columns per block (8 KB of float2 in LDS)
#define ROWS_PER_BLOCK 8      // rows per block
#define R0F            0.1f

typedef float v4f __attribute__((ext_vector_type(4)));  // native clang vector

__global__ __launch_bounds__(THREADS) void radiusgraph_edge_kernel(
    const float* __restrict__ x,
    float* __restrict__ dist_out,
    float* __restrict__ mask_out,
    int N) {
  __shared__ __align__(16) float sx[TILE_COLS * 2];  // column tile positions

  const int tid  = threadIdx.x;
  const int col0 = blockIdx.x * TILE_COLS;
  const int row0 = blockIdx.y * ROWS_PER_BLOCK;

  const bool full_cols = (col0 + TILE_COLS) <= N;

  if (full_cols) {
    // ---- Async-copy the 8 KB column tile (1024 x float2) into LDS.
    // Each of 256 lanes moves 16 B twice: 2 * 4 KB = 8 KB.
    unsigned lds_base = (unsigned)(unsigned long long)
        (__attribute__((address_space(3))) unsigned char*)sx;
    unsigned lds0 = lds_base + (unsigned)tid * 16u;
    unsigned off0 = (unsigned)col0 * 8u + (unsigned)tid * 16u;
    asm volatile("global_load_async_to_lds_b128 %0, %1, %2"
                 :: "v"(lds0), "v"(off0), "s"(x) : "memory");
    asm volatile("global_load_async_to_lds_b128 %0, %1, %2"
                 :: "v"(lds0 + 4096u), "v"(off0 + 4096u), "s"(x) : "memory");
    asm volatile("s_wait_asynccnt 0" ::: "memory");
  } else {
    // Generic (partial-tile) fallback: bounds-checked synchronous fill.
    for (int c = tid; c < TILE_COLS; c += THREADS) {
      int j = col0 + c;
      float vx = 0.0f, vy = 0.0f;
      if (j < N) { vx = x[2 * j]; vy = x[2 * j + 1]; }
      sx[2 * c]     = vx;
      sx[2 * c + 1] = vy;
    }
  }
  __syncthreads();

  // Each thread owns 4 consecutive columns: two b128 LDS reads.
  const v4f* s4 = (const v4f*)sx;
  const v4f c01 = s4[tid * 2 + 0];   // xj0.x xj0.y xj1.x xj1.y
  const v4f c23 = s4[tid * 2 + 1];   // xj2.x xj2.y xj3.x xj3.y
  const float jx[4] = {c01.x, c01.z, c23.x, c23.z};
  const float jy[4] = {c01.y, c01.w, c23.y, c23.w};
  const int j0 = col0 + tid * 4;

#pragma unroll
  for (int r = 0; r < ROWS_PER_BLOCK; ++r) {
    const int i = row0 + r;
    if (i >= N) break;
    const float xi = x[2 * i];       // uniform per block-row -> scalar loads
    const float yi = x[2 * i + 1];

    float dv[4], mv[4];
#pragma unroll
    for (int k = 0; k < 4; ++k) {
      float dx = jx[k] - xi;
      dx -= rintf(dx);               // v_rndne_f32: minimum image, L = 1
      float dy = jy[k] - yi;
      dy -= rintf(dy);
      float d2 = fmaf(dx, dx, dy * dy);
      float dd = __builtin_amdgcn_sqrtf(d2);   // single v_sqrt_f32
      dv[k] = dd;
      mv[k] = ((dd < R0F) && ((j0 + k) != i)) ? 1.0f : 0.0f;
    }

    const size_t base = (size_t)i * (size_t)N + (size_t)j0;
    if (j0 + 4 <= N) {
      v4f d4 = {dv[0], dv[1], dv[2], dv[3]};
      v4f m4 = {mv[0], mv[1], mv[2], mv[3]};
      // Streaming outputs: nontemporal so 512 MB of dead lines skip L2 reuse.
      __builtin_nontemporal_store(d4, (v4f*)(dist_out + base));
      __builtin_nontemporal_store(m4, (v4f*)(mask_out + base));
    } else {
      for (int k = 0; k < 4 && (j0 + k) < N; ++k) {
        dist_out[base + k] = dv[k];
        mask_out[base + k] = mv[k];
      }
    }
  }
}

extern "C" void kernel_launch(void* const* d_in, const int* in_sizes, int n_in,
                              void* d_out, int out_size, void* d_ws, size_t ws_size,
                              hipStream_t stream) {
  (void)n_in; (void)d_ws; (void)ws_size; (void)out_size;
  const float* x = (const float*)d_in[0];
  const int N = in_sizes[0] / 2;                 // x is [N][2] float32

  float* dist = (float*)d_out;
  float* mask = dist + (size_t)N * (size_t)N;

  dim3 grid((N + TILE_COLS - 1) / TILE_COLS,
            (N + ROWS_PER_BLOCK - 1) / ROWS_PER_BLOCK);
  hipLaunchKernelGGL(radiusgraph_edge_kernel, grid, dim3(THREADS), 0, stream,
                     x, dist, mask, N);
}